// CustomGAE_38422777430705
// MI455X (gfx1250) — compile-verified
//
#include <hip/hip_runtime.h>

// GAE: value = obs@W + b ; next_value = next_obs@W + b
//      td0  = r + g*nd*next_value - value ; coef = g*l*nd
//      A_t  = td0_t + coef_t * A_{t+1}  (reverse scan over T)
//      out  = [advantage | advantage + value]

#define GAMMA 0.99f
#define LMBDA 0.95f

typedef __attribute__((ext_vector_type(2))) float v2f;
typedef __attribute__((ext_vector_type(8))) float v8f;

constexpr int Bsz = 512;
constexpr int Tsz = 2048;
constexpr int Dsz = 64;

// ---------------------------------------------------------------------------
// Phase 1: fp32 WMMA GEMV. One wave per 16-row tile of the flattened
// [B*T, 64] obs/next_obs matrices. 16 K-chunks of V_WMMA_F32_16X16X4_F32,
// B-matrix = W broadcast across all 16 N columns, so D column 0..15 all hold
// the dot products; lane 0 holds rows 0-7, lane 16 holds rows 8-15.
// ---------------------------------------------------------------------------
__global__ __launch_bounds__(256) void gae_value_gemv(
    const float* __restrict__ obs, const float* __restrict__ next_obs,
    const float* __restrict__ W, const float* __restrict__ bias,
    float* __restrict__ value_ws, float* __restrict__ next_value_ws)
{
    const int wave = (int)((blockIdx.x * blockDim.x + threadIdx.x) >> 5);
    const int lane = (int)(threadIdx.x & 31);
    const int m    = lane & 15;   // A-matrix row within tile
    const int half = lane >> 4;   // half-wave selects K pair {0,1} vs {2,3}

    // Preload B-tiles: chunk c covers K=4c..4c+3; this lane supplies
    // B[K=4c+2*half][*] and B[K=4c+2*half+1][*], replicated over N.
    v2f wreg[16];
#pragma unroll
    for (int c = 0; c < 16; ++c)
        wreg[c] = *(const v2f*)(W + 4 * c + 2 * half);

    const float* ap = obs      + (size_t)(wave * 16 + m) * Dsz + 2 * half;
    const float* np = next_obs + (size_t)(wave * 16 + m) * Dsz + 2 * half;

    v8f acc  = {};
    v8f nacc = {};
#pragma unroll
    for (int c = 0; c < 16; ++c) {
        v2f a  = *(const v2f*)(ap + 4 * c);
        v2f na = *(const v2f*)(np + 4 * c);
        acc  = __builtin_amdgcn_wmma_f32_16x16x4_f32(false, a,  false, wreg[c],
                                                     (short)0, acc,  false, false);
        nacc = __builtin_amdgcn_wmma_f32_16x16x4_f32(false, na, false, wreg[c],
                                                     (short)0, nacc, false, false);
    }

    // D layout (32-bit C/D 16x16): VGPR r, lanes 0-15 -> M=r, lanes 16-31 -> M=r+8.
    // All N columns are identical, so lanes 0 and 16 carry 8 consecutive rows each.
    if (m == 0) {
        const float bv = bias[0];
        float* vp = value_ws      + (size_t)wave * 16 + half * 8;
        float* xp = next_value_ws + (size_t)wave * 16 + half * 8;
        float4 v0 = { acc[0] + bv,  acc[1] + bv,  acc[2] + bv,  acc[3] + bv };
        float4 v1 = { acc[4] + bv,  acc[5] + bv,  acc[6] + bv,  acc[7] + bv };
        float4 n0 = { nacc[0] + bv, nacc[1] + bv, nacc[2] + bv, nacc[3] + bv };
        float4 n1 = { nacc[4] + bv, nacc[5] + bv, nacc[6] + bv, nacc[7] + bv };
        *(float4*)(vp)     = v0;
        *(float4*)(vp + 4) = v1;
        *(float4*)(xp)     = n0;
        *(float4*)(xp + 4) = n1;
    }
}

// ---------------------------------------------------------------------------
// Phase 2: parallel reverse linear recurrence. One block per batch row.
// Thread j folds t in [8j, 8j+8) into an affine map A_start = u + v*A_end,
// Hillis-Steele right-scan over the 256 maps in LDS, then replay with carry.
// ---------------------------------------------------------------------------
__global__ __launch_bounds__(256) void gae_scan(
    const float* __restrict__ value_ws, const float* __restrict__ next_value_ws,
    const float* __restrict__ reward, const int* __restrict__ done,
    float* __restrict__ adv_out, float* __restrict__ vt_out)
{
    __shared__ float su[256];
    __shared__ float sv[256];
    const int b    = (int)blockIdx.x;
    const int j    = (int)threadIdx.x;
    const int base = b * Tsz + j * 8;

    __align__(16) float val[8], nxt[8], rw[8];
    __align__(16) int   dn[8];
    *(float4*)&val[0] = *(const float4*)(value_ws + base);
    *(float4*)&val[4] = *(const float4*)(value_ws + base + 4);
    *(float4*)&nxt[0] = *(const float4*)(next_value_ws + base);
    *(float4*)&nxt[4] = *(const float4*)(next_value_ws + base + 4);
    *(float4*)&rw[0]  = *(const float4*)(reward + base);
    *(float4*)&rw[4]  = *(const float4*)(reward + base + 4);
    *(int4*)&dn[0]    = *(const int4*)(done + base);
    *(int4*)&dn[4]    = *(const int4*)(done + base + 4);

    float td[8], cf[8];
#pragma unroll
    for (int k = 0; k < 8; ++k) {
        const float nd = dn[k] ? 0.0f : 1.0f;
        td[k] = rw[k] + GAMMA * nd * nxt[k] - val[k];
        cf[k] = (GAMMA * LMBDA) * nd;
    }

    // Local fold (right-to-left): chunk map (u, v) with A_start = u + v * A_end.
    float u = 0.0f, v = 1.0f;
#pragma unroll
    for (int k = 7; k >= 0; --k) { u = td[k] + cf[k] * u; v *= cf[k]; }
    su[j] = u; sv[j] = v;
    __syncthreads();

    // Inclusive right-scan of affine compositions: S_j = f_j ∘ f_{j+1} ∘ ... ∘ f_255
#pragma unroll
    for (int off = 1; off < 256; off <<= 1) {
        const float u1 = su[j], v1 = sv[j];
        float u2 = 0.0f, v2 = 1.0f;           // identity beyond the end
        if (j + off < 256) { u2 = su[j + off]; v2 = sv[j + off]; }
        __syncthreads();
        su[j] = u1 + v1 * u2;
        sv[j] = v1 * v2;
        __syncthreads();
    }

    // Carry entering this chunk from the right = S_{j+1}(0)
    const float carry = (j + 1 < 256) ? su[j + 1] : 0.0f;

    float A = carry;
    __align__(16) float adv[8], vt[8];
#pragma unroll
    for (int k = 7; k >= 0; --k) {
        A = td[k] + cf[k] * A;
        adv[k] = A;
        vt[k]  = A + val[k];
    }
    *(float4*)(adv_out + base)     = *(float4*)&adv[0];
    *(float4*)(adv_out + base + 4) = *(float4*)&adv[4];
    *(float4*)(vt_out + base)      = *(float4*)&vt[0];
    *(float4*)(vt_out + base + 4)  = *(float4*)&vt[4];
}

extern "C" void kernel_launch(void* const* d_in, const int* in_sizes, int n_in,
                              void* d_out, int out_size, void* d_ws, size_t ws_size,
                              hipStream_t stream)
{
    (void)in_sizes; (void)n_in; (void)out_size; (void)ws_size;

    const float* obs     = (const float*)d_in[0];
    const float* nobs    = (const float*)d_in[1];
    const float* reward  = (const float*)d_in[2];
    const int*   done    = (const int*)d_in[3];
    const float* W       = (const float*)d_in[4];
    const float* bias    = (const float*)d_in[5];

    float* value_ws      = (float*)d_ws;                       // [B*T]
    float* next_value_ws = value_ws + (size_t)Bsz * Tsz;       // [B*T]  (8 MB total)

    float* adv = (float*)d_out;                                // advantage [B,T,1]
    float* vt  = adv + (size_t)Bsz * Tsz;                      // value_target [B,T,1]

    // Phase 1: (B*T)/16 = 65536 waves, 8 waves per 256-thread block.
    const int tiles  = (Bsz * Tsz) / 16;
    const int blocks = tiles / 8;
    gae_value_gemv<<<blocks, 256, 0, stream>>>(obs, nobs, W, bias,
                                               value_ws, next_value_ws);

    // Phase 2: one block per batch row.
    gae_scan<<<Bsz, 256, 0, stream>>>(value_ws, next_value_ws, reward, done,
                                      adv, vt);
}